// MHCLayerPyTorchBwd_44023414784000
// MI455X (gfx1250) — compile-verified
//
#include <hip/hip_runtime.h>

typedef float v2f __attribute__((ext_vector_type(2)));
typedef float v4f __attribute__((ext_vector_type(4)));
typedef float v8f __attribute__((ext_vector_type(8)));
typedef unsigned int u32x4 __attribute__((ext_vector_type(4)));
typedef int i32x4 __attribute__((ext_vector_type(4)));
typedef int i32x8 __attribute__((ext_vector_type(8)));

#define B_ROWS 4096
#define C_DIM  4096
#define NC     16384      // 4 * 4096 flattened row length
#define NPROJ  24         // 4 pre + 4 post + 16 res
#define NPAD   32         // padded projection count
#define SPLITK 4          // split-K factor for the GEMM
#define KCHUNK 256        // K elements staged per TDM transfer
#define NCHUNK 16         // (NC/SPLITK)/KCHUNK
#define LROW   260        // KCHUNK + 4 dword pad (bank-conflict-free)
#define SINK_EPS 1e-8f
#define RMS_EPS  1e-5f
#define N_ITERS  20

// ---------------------------------------------------------------------------
// Kernel 0: pack PhiT[k][j] = alpha_* * phi_*[j][k], j in [0,32) (24 real,
// 8 zero pad). Folding alpha here makes the GEMM output pre-scaled.
// ---------------------------------------------------------------------------
__global__ __launch_bounds__(256) void pack_phi(
    const float* __restrict__ phi_pre, const float* __restrict__ phi_post,
    const float* __restrict__ phi_res, const float* __restrict__ alpha_pre,
    const float* __restrict__ alpha_post, const float* __restrict__ alpha_res,
    float* __restrict__ phiT) {
  int idx = blockIdx.x * 256 + threadIdx.x;   // 0 .. 32*16384
  int j = idx / NC;                           // projection index
  int k = idx % NC;                           // flat feature index
  float v = 0.0f;
  if (j < 4)        v = alpha_pre[0]  * phi_pre [(size_t)j * NC + k];
  else if (j < 8)   v = alpha_post[0] * phi_post[(size_t)(j - 4) * NC + k];
  else if (j < 24)  v = alpha_res[0]  * phi_res [(size_t)(j - 8) * NC + k];
  phiT[(size_t)k * NPAD + j] = v;
}

// ---------------------------------------------------------------------------
// TDM: issue a 2D tile load (KCHUNK floats x 16 rows, row stride NC floats)
// from global into LDS at lds_off, with 4 dwords of LDS padding appended per
// 256-dword row (D# pad_interval=7 -> 256 dw, pad_amount=3 -> 4 dw).
// D# bit layout per CDNA5 ISA 8.3/8.4. Groups 2/3 (and the trailing group
// of the 6-arg clang-23 builtin) are zero: 2D tile needs only groups 0/1.
// ---------------------------------------------------------------------------
__device__ __forceinline__ void tdm_load_tile(unsigned int lds_off,
                                              const float* gptr) {
  unsigned long long ga = (unsigned long long)(uintptr_t)gptr;
  u32x4 g0;
  g0[0] = 1u;                                   // count=1, user mode
  g0[1] = lds_off;                              // lds_addr
  g0[2] = (unsigned int)ga;                     // global_addr[31:0]
  g0[3] = (unsigned int)(ga >> 32) | (2u << 30);// global_addr[56:32] | type=2
  i32x8 g1;
  g1[0] = (int)((2u << 16)      // data_size = 4 bytes
              | (1u << 20)      // pad_enable
              | (7u << 22)      // pad_interval: 256 dwords
              | (3u << 25));    // pad_amount: 4 dwords
  g1[1] = (int)((unsigned)NC << 16);      // tensor_dim0 lo16 @ [63:48]
  g1[2] = (int)(16u << 16);               // tensor_dim1 lo16 @ [95:80]
  g1[3] = (int)((unsigned)KCHUNK << 16);  // tile_dim0 @ [127:112]
  g1[4] = 16;                             // tile_dim1 @ [143:128]
  g1[5] = NC;                             // tensor_dim0_stride lo32
  g1[6] = 0;
  g1[7] = 0;
  i32x4 gz4 = {0, 0, 0, 0};
  i32x8 gz8 = {0, 0, 0, 0, 0, 0, 0, 0};
  __builtin_amdgcn_tensor_load_to_lds(g0, g1, gz4, gz4, gz8, 0);
}

__device__ __forceinline__ unsigned int lds_addr_of(const void* p) {
  // generic LDS addresses carry the LDS byte offset in bits [31:0]
  return (unsigned int)(uintptr_t)p;
}

// ---------------------------------------------------------------------------
// Kernel 1: P[b, 0..31] = sum_k x[b,k] * PhiT[k, 0..31]  (raw, pre-scaled)
// One block = 16 rows of B, one split-K slice (4096 K). The 16x256 A tiles
// are DMA'd into double-buffered LDS by the Tensor Data Mover; 8 waves each
// take 32 K of every chunk and accumulate [16x32] with
// v_wmma_f32_16x16x4_f32 (exact fp32); tiles reduce across waves via LDS.
//
// ISA lane layout (32-bit 16x4 A): lane l holds A[m = l&15][k0 + 2*(l>>4)+v]
// for vgpr v in {0,1}; B mirrors it: B[k0 + 2*(l>>4)+v][n = l&15].
// C/D: vgpr r, lane l -> (M = r + 8*(l>>4), N = l&15).
// ---------------------------------------------------------------------------
__global__ __launch_bounds__(256) void proj_gemm(
    const float* __restrict__ x, const float* __restrict__ phiT,
    float* __restrict__ pproj) {
  __shared__ float lbuf[2][16 * LROW];        // 2 x 16.25 KB staged A tiles
  __shared__ float red[8][16][NPAD];          // 16 KB cross-wave reduce

  const int tid  = threadIdx.x;
  const int lane = tid & 31;
  const int wave = tid >> 5;
  const int s    = blockIdx.x >> 8;           // split-K slice 0..3
  const int g    = blockIdx.x & 255;          // row-group 0..255
  const int b0   = g * 16;

  const int m  = lane & 15;                   // A row within tile
  const int hk = (lane >> 4) << 1;            // 0 or 2: K-pair select
  const int n  = lane & 15;                   // B/C column within tile

  const int kslice = s * (NC / SPLITK);       // base K of this slice
  const float* xbase = x + (size_t)b0 * NC + kslice;

  // LDS read pointers for this lane's A fragments (bank-conflict-free)
  const float* ap0 = &lbuf[0][m * LROW + wave * 32 + hk];
  const float* ap1 = &lbuf[1][m * LROW + wave * 32 + hk];

  v8f acc0 = {};   // columns 0..15
  v8f acc1 = {};   // columns 16..31

  // prologue: DMA chunk 0
  if (wave == 0)
    tdm_load_tile(lds_addr_of(&lbuf[0][0]), xbase);

  for (int c = 0; c < NCHUNK; ++c) {
    if (wave == 0)
      __builtin_amdgcn_s_wait_tensorcnt(0);   // chunk c landed in LDS
    __syncthreads();

    // overlap: DMA chunk c+1 into the other buffer while computing on c
    if (wave == 0 && (c + 1) < NCHUNK)
      tdm_load_tile(lds_addr_of(&lbuf[(c + 1) & 1][0]),
                    xbase + (c + 1) * KCHUNK);

    const float* ab = (c & 1) ? ap1 : ap0;
    const float* bb = phiT +
        (size_t)(kslice + c * KCHUNK + wave * 32 + hk) * NPAD + n;

#pragma unroll
    for (int t = 0; t < 8; ++t) {             // 8 k-steps of 4 per chunk/wave
      v2f a = *(const v2f*)(ab + t * 4);      // ds_load_b64, const offset
      v2f bf0, bf1;
      bf0.x = bb[t * 128];        bf0.y = bb[t * 128 + 32];
      bf1.x = bb[t * 128 + 16];   bf1.y = bb[t * 128 + 48];
      acc0 = __builtin_amdgcn_wmma_f32_16x16x4_f32(
          false, a, false, bf0, (short)0, acc0, false, false);
      acc1 = __builtin_amdgcn_wmma_f32_16x16x4_f32(
          false, a, false, bf1, (short)0, acc1, false, false);
    }
    __syncthreads();                          // buffer reuse fence
  }

  // spill per-wave tiles to LDS in (m, n) layout
  const int mbase = (lane >> 4) * 8;
#pragma unroll
  for (int r = 0; r < 8; ++r) {
    red[wave][mbase + r][n]      = acc0[r];
    red[wave][mbase + r][n + 16] = acc1[r];
  }
  __syncthreads();

  // cross-wave reduction: 512 outputs, 2 per thread
  for (int e = tid; e < 16 * NPAD; e += 256) {
    int mm = e / NPAD, nn = e % NPAD;
    float sum = 0.0f;
#pragma unroll
    for (int w = 0; w < 8; ++w) sum += red[w][mm][nn];
    pproj[((size_t)s * B_ROWS + (b0 + mm)) * NPAD + nn] = sum;
  }
}

// ---------------------------------------------------------------------------
// Kernel 2: per-row finalize. One block per row b. The 16K-float row lives in
// registers (64 VGPRs/thread, coalesced b128 traffic), Gram matrix via wave32
// shuffles + tiny LDS, lane-0 does softmax/sigmoid/exp + 20 Sinkhorn iters,
// then the output streams are produced straight from registers.
// ---------------------------------------------------------------------------
__device__ inline float vsum4(v4f v) { return v.x + v.y + v.z + v.w; }

__global__ __launch_bounds__(256) void finalize(
    const float* __restrict__ x, const float* __restrict__ w,
    const float* __restrict__ b_pre, const float* __restrict__ b_post,
    const float* __restrict__ b_res, const float* __restrict__ pproj,
    float* __restrict__ out) {
  __shared__ float gred[8][10];
  __shared__ float coef[24];      // p[4], a[4] = H_post*inv, M[16]

  const int tid = threadIdx.x;
  const int b   = blockIdx.x;

  // ---- load row into registers: stream n, chunk u, 4 floats per thread ----
  const v4f* xr4 = (const v4f*)(x + (size_t)b * NC);
  v4f xs[4][4];
#pragma unroll
  for (int nS = 0; nS < 4; ++nS)
#pragma unroll
    for (int u = 0; u < 4; ++u)
      xs[nS][u] = xr4[nS * 1024 + u * 256 + tid];

  // ---- Gram partials: 10 unique pair sums ----
  float gacc[10];
#pragma unroll
  for (int i = 0; i < 10; ++i) gacc[i] = 0.0f;
#pragma unroll
  for (int u = 0; u < 4; ++u) {
    v4f A = xs[0][u], Bv = xs[1][u], Cv = xs[2][u], Dv = xs[3][u];
    gacc[0] += vsum4(A * A);   gacc[1] += vsum4(A * Bv);
    gacc[2] += vsum4(A * Cv);  gacc[3] += vsum4(A * Dv);
    gacc[4] += vsum4(Bv * Bv); gacc[5] += vsum4(Bv * Cv);
    gacc[6] += vsum4(Bv * Dv); gacc[7] += vsum4(Cv * Cv);
    gacc[8] += vsum4(Cv * Dv); gacc[9] += vsum4(Dv * Dv);
  }
#pragma unroll
  for (int i = 0; i < 10; ++i)
    for (int off = 16; off > 0; off >>= 1)
      gacc[i] += __shfl_down(gacc[i], off, 32);
  if ((tid & 31) == 0) {
#pragma unroll
    for (int i = 0; i < 10; ++i) gred[tid >> 5][i] = gacc[i];
  }
  __syncthreads();

  // ---- scalar math on thread 0 ----
  if (tid == 0) {
    float gs[10];
#pragma unroll
    for (int i = 0; i < 10; ++i) {
      float ssum = 0.0f;
#pragma unroll
      for (int wv = 0; wv < 8; ++wv) ssum += gred[wv][i];
      gs[i] = ssum;
    }
    float G[4][4];
    G[0][0] = gs[0]; G[0][1] = G[1][0] = gs[1];
    G[0][2] = G[2][0] = gs[2]; G[0][3] = G[3][0] = gs[3];
    G[1][1] = gs[4]; G[1][2] = G[2][1] = gs[5]; G[1][3] = G[3][1] = gs[6];
    G[2][2] = gs[7]; G[2][3] = G[3][2] = gs[8]; G[3][3] = gs[9];

    float sumsq = G[0][0] + G[1][1] + G[2][2] + G[3][3];
    float inv_rms = 1.0f / sqrtf(sumsq / (float)NC + SINK_EPS);

    // sum the SPLITK partial projections (alpha already folded in)
    float pp[NPROJ];
#pragma unroll
    for (int j = 0; j < NPROJ; ++j) {
      float ssum = 0.0f;
#pragma unroll
      for (int sK = 0; sK < SPLITK; ++sK)
        ssum += pproj[((size_t)sK * B_ROWS + b) * NPAD + j];
      pp[j] = ssum;
    }

    // H_pre: softmax over 4
    float hp[4];
#pragma unroll
    for (int nS = 0; nS < 4; ++nS) hp[nS] = pp[nS] * inv_rms + b_pre[nS];
    float mx = fmaxf(fmaxf(hp[0], hp[1]), fmaxf(hp[2], hp[3]));
    float se = 0.0f;
#pragma unroll
    for (int nS = 0; nS < 4; ++nS) { hp[nS] = expf(hp[nS] - mx); se += hp[nS]; }
    float p[4];
#pragma unroll
    for (int nS = 0; nS < 4; ++nS) p[nS] = hp[nS] / se;

    // H_post: 2*sigmoid
    float hq[4];
#pragma unroll
    for (int nS = 0; nS < 4; ++nS) {
      float v = pp[4 + nS] * inv_rms + b_post[nS];
      hq[nS] = 2.0f / (1.0f + expf(-v));
    }

    // Sinkhorn on exp(H_res_raw), 20 iterations of row/col normalization
    float M[4][4];
#pragma unroll
    for (int i = 0; i < 4; ++i)
#pragma unroll
      for (int j = 0; j < 4; ++j)
        M[i][j] = expf(pp[8 + i * 4 + j] * inv_rms + b_res[i * 4 + j]);
    for (int it = 0; it < N_ITERS; ++it) {
#pragma unroll
      for (int i = 0; i < 4; ++i) {
        float rs = M[i][0] + M[i][1] + M[i][2] + M[i][3] + SINK_EPS;
        float r = 1.0f / rs;
#pragma unroll
        for (int j = 0; j < 4; ++j) M[i][j] *= r;
      }
#pragma unroll
      for (int j = 0; j < 4; ++j) {
        float cs = M[0][j] + M[1][j] + M[2][j] + M[3][j] + SINK_EPS;
        float r = 1.0f / cs;
#pragma unroll
        for (int i = 0; i < 4; ++i) M[i][j] *= r;
      }
    }

    // RMSNorm inverse of x_agg via Gram: mean(x_agg^2) = p^T G p / C
    float msq = 0.0f;
#pragma unroll
    for (int nS = 0; nS < 4; ++nS)
#pragma unroll
      for (int mS = 0; mS < 4; ++mS) msq += p[nS] * p[mS] * G[nS][mS];
    float inv = rsqrtf(msq / (float)C_DIM + RMS_EPS);

#pragma unroll
    for (int nS = 0; nS < 4; ++nS) coef[nS] = p[nS];
#pragma unroll
    for (int i = 0; i < 4; ++i) coef[4 + i] = hq[i] * inv;
#pragma unroll
    for (int i = 0; i < 4; ++i)
#pragma unroll
      for (int j = 0; j < 4; ++j) coef[8 + i * 4 + j] = M[i][j];
  }
  __syncthreads();

  const float p0 = coef[0], p1 = coef[1], p2 = coef[2], p3 = coef[3];
  const float a0 = coef[4], a1 = coef[5], a2 = coef[6], a3 = coef[7];
  const float m00 = coef[8],  m01 = coef[9],  m02 = coef[10], m03 = coef[11];
  const float m10 = coef[12], m11 = coef[13], m12 = coef[14], m13 = coef[15];
  const float m20 = coef[16], m21 = coef[17], m22 = coef[18], m23 = coef[19];
  const float m30 = coef[20], m31 = coef[21], m32 = coef[22], m33 = coef[23];

  v4f* o4 = (v4f*)(out + (size_t)b * NC);
  const v4f* w4 = (const v4f*)w;
#pragma unroll
  for (int u = 0; u < 4; ++u) {
    v4f A = xs[0][u], Bv = xs[1][u], Cv = xs[2][u], Dv = xs[3][u];
    v4f wv  = w4[u * 256 + tid];
    v4f agg = p0 * A + p1 * Bv + p2 * Cv + p3 * Dv;
    v4f yw  = wv * agg;
    const int off = u * 256 + tid;
    o4[0 * 1024 + off] = a0 * yw + m00 * A + m01 * Bv + m02 * Cv + m03 * Dv;
    o4[1 * 1024 + off] = a1 * yw + m10 * A + m11 * Bv + m12 * Cv + m13 * Dv;
    o4[2 * 1024 + off] = a2 * yw + m20 * A + m21 * Bv + m22 * Cv + m23 * Dv;
    o4[3 * 1024 + off] = a3 * yw + m30 * A + m31 * Bv + m32 * Cv + m33 * Dv;
  }
}

// ---------------------------------------------------------------------------
extern "C" void kernel_launch(void* const* d_in, const int* in_sizes, int n_in,
                              void* d_out, int out_size, void* d_ws,
                              size_t ws_size, hipStream_t stream) {
  (void)in_sizes; (void)n_in; (void)out_size; (void)ws_size;
  const float* x        = (const float*)d_in[0];
  const float* wgt      = (const float*)d_in[1];
  const float* phi_pre  = (const float*)d_in[2];
  const float* phi_post = (const float*)d_in[3];
  const float* phi_res  = (const float*)d_in[4];
  const float* b_pre    = (const float*)d_in[5];
  const float* b_post   = (const float*)d_in[6];
  const float* b_res    = (const float*)d_in[7];
  const float* a_pre    = (const float*)d_in[8];
  const float* a_post   = (const float*)d_in[9];
  const float* a_res    = (const float*)d_in[10];

  float* phiT  = (float*)d_ws;                         // [NC][32]  = 2 MB
  float* pproj = phiT + (size_t)NC * NPAD;             // [SPLITK][B][32] = 2 MB
  float* out   = (float*)d_out;

  pack_phi<<<(NC * NPAD) / 256, 256, 0, stream>>>(
      phi_pre, phi_post, phi_res, a_pre, a_post, a_res, phiT);
  proj_gemm<<<SPLITK * (B_ROWS / 16), 256, 0, stream>>>(x, phiT, pproj);
  finalize<<<B_ROWS, 256, 0, stream>>>(x, wgt, b_pre, b_post, b_res, pproj, out);
}